// AsyncGraph_88038239633883
// MI455X (gfx1250) — compile-verified
//
#include <hip/hip_runtime.h>
#include <math.h>
#include <stdint.h>

typedef __attribute__((ext_vector_type(2))) float v2f;
typedef __attribute__((ext_vector_type(8))) float v8f;

#define DEV __device__ __forceinline__

// problem constants
constexpr int Bn = 8, Pn = 16, Wn = 128, Dn = 256;

DEV v8f vzero() { v8f c; for (int i = 0; i < 8; ++i) c[i] = 0.f; return c; }

DEV v8f wmma4(v2f a, v2f b, v8f c) {
  return __builtin_amdgcn_wmma_f32_16x16x4_f32(false, a, false, b, (short)0, c,
                                               false, false);
}

// Async copy of `n16` 16-byte chunks from global `g` to LDS `l`, cooperatively
// across the whole block; leaves data visible after the barrier.
// Uses CDNA5 GLOBAL_LOAD_ASYNC_TO_LDS_B128 (ASYNCcnt) — no VGPR data movement.
DEV void async_stage(const float* g, float* l, int n16) {
  uint64_t gbase = (uint64_t)(uintptr_t)g;
  uint32_t lbase = (uint32_t)(uintptr_t)l;  // low 32 bits = LDS offset
  for (int i = threadIdx.x; i < n16; i += blockDim.x) {
    uint64_t ga = gbase + (uint64_t)i * 16u;
    uint32_t la = lbase + (uint32_t)i * 16u;
    asm volatile("global_load_async_to_lds_b128 %0, %1, off"
                 :: "v"(la), "v"(ga) : "memory");
  }
  asm volatile("s_wait_asynccnt 0x0" ::: "memory");
  __syncthreads();
}

// C(16x16) += A(16xK, row-major, row stride lda) * Bw(Kx16 slice of KxN weight, row stride ldb)
DEV v8f gemm_w(const float* A, int lda, const float* Bw, int ldb, int K, v8f c) {
  int lane = threadIdx.x & 31;
  int idx = lane & 15, hi = lane >> 4;
  for (int k0 = 0; k0 < K; k0 += 4) {
    int ka = k0 + 2 * hi;
    v2f a; a.x = A[idx * lda + ka];      a.y = A[idx * lda + ka + 1];
    v2f b; b.x = Bw[ka * ldb + idx];     b.y = Bw[(ka + 1) * ldb + idx];
    c = wmma4(a, b, c);
  }
  return c;
}

// C(16x16) += A(16xK, row stride lda) * Bt(16xK, row stride ldb)^T   (row-dot form)
DEV v8f gemm_nt(const float* A, long lda, const float* Bt, long ldb, int K, v8f c) {
  int lane = threadIdx.x & 31;
  int idx = lane & 15, hi = lane >> 4;
  for (int k0 = 0; k0 < K; k0 += 4) {
    int ka = k0 + 2 * hi;
    v2f a; a.x = A[idx * lda + ka];  a.y = A[idx * lda + ka + 1];
    v2f b; b.x = Bt[idx * ldb + ka]; b.y = Bt[idx * ldb + ka + 1];
    c = wmma4(a, b, c);
  }
  return c;
}

// ---------------- K0: fold mean-over-A weights; zero edge accumulator ----------
__global__ __launch_bounds__(256) void k_prep(const float* w_node, const float* b_node,
                                              const float* w_a, const float* b_a,
                                              const float* w_edge, const float* b_edge,
                                              float* wn, float* vae, float* sc,
                                              float* edges) {
  __shared__ float we[256];
  __shared__ float red[256];
  int t = threadIdx.x;
  float s1 = 0.f, s2 = 0.f;
  for (int a = 0; a < 16; ++a) { s1 += w_node[t * 16 + a]; s2 += w_edge[t * 16 + a]; }
  wn[t] = s1 * (1.f / 16.f);
  we[t] = s2 * (1.f / 16.f);
  __syncthreads();
  float v = 0.f;
  for (int k = 0; k < 256; ++k) v += w_a[t * 256 + k] * we[k];
  vae[t] = v;
  // eb = dot(b_a, we) + mean(b_edge)
  red[t] = b_a[t] * we[t] + ((t < 16) ? b_edge[t] * (1.f / 16.f) : 0.f);
  __syncthreads();
  for (int o = 128; o > 0; o >>= 1) { if (t < o) red[t] += red[t + o]; __syncthreads(); }
  if (t == 0) sc[1] = red[0];
  __syncthreads();
  red[t] = (t < 16) ? b_node[t] * (1.f / 16.f) : 0.f;
  __syncthreads();
  for (int o = 128; o > 0; o >>= 1) { if (t < o) red[t] += red[t + o]; __syncthreads(); }
  if (t == 0) sc[0] = red[0];
  for (int i = t; i < Bn * Pn * Pn * 4; i += 256) edges[i] = 0.f;
}

// ---------------- K1: fused projection GEMM (Bp, Cp, mreal, mdel) --------------
__global__ __launch_bounds__(256) void k_proj(const float* fea,
                                              const float* w_b, const float* b_b,
                                              const float* w_c, const float* b_c,
                                              const float* w_r, const float* b_r,
                                              const float* w_d, const float* b_d,
                                              float* Bp, float* Cp, float* Mr,
                                              float* Md) {
  __shared__ float As[16 * 256];
  long rt = blockIdx.x;  // 1024 row tiles of 16 rows over (b,p,w)
  async_stage(fea + rt * 16 * 256, As, 1024);
  int wave = threadIdx.x >> 5;
  int job = blockIdx.y * 8 + wave;  // gridDim.y = 6 -> jobs 0..47
  const float *Wm, *bias; float* out; int ldo, ncol;
  if (job < 16)      { Wm = w_b; bias = b_b; out = Bp; ldo = 256; ncol = job * 16; }
  else if (job < 32) { Wm = w_c; bias = b_c; out = Cp; ldo = 256; ncol = (job - 16) * 16; }
  else if (job < 40) { Wm = w_r; bias = b_r; out = Mr; ldo = 128; ncol = (job - 32) * 16; }
  else               { Wm = w_d; bias = b_d; out = Md; ldo = 128; ncol = (job - 40) * 16; }
  v8f c = vzero();
  c = gemm_w(As, 256, Wm + ncol, ldo, 256, c);
  int lane = threadIdx.x & 31, n = lane & 15, hi = lane >> 4;
  float bv = bias[ncol + n];
  float* o = out + rt * 16 * (long)ldo + ncol;
  for (int r = 0; r < 8; ++r) o[(long)(r + 8 * hi) * ldo + n] = c[r] + bv;
}

// ---------------- K2: e[b,p,w] and node_cost[b,p] ------------------------------
__global__ __launch_bounds__(256) void k_e_node(const float* fea, const float* wn,
                                                const float* vae, const float* sc,
                                                float* ebuf, float* nc) {
  int bp = blockIdx.x;
  int wave = threadIdx.x >> 5, lane = threadIdx.x & 31;
  __shared__ float nred[8];
  const float* base = fea + (long)bp * Wn * Dn;
  float nsum = 0.f;
  for (int w = wave * 16; w < wave * 16 + 16; ++w) {
    const float* row = base + (long)w * Dn;
    float de = 0.f, dn = 0.f;
    for (int d = lane; d < 256; d += 32) {
      float f = row[d];
      de += f * vae[d];
      dn += f * wn[d];
    }
    for (int o = 16; o; o >>= 1) {
      de += __shfl_xor(de, o, 32);
      dn += __shfl_xor(dn, o, 32);
    }
    if (lane == 0) { ebuf[(long)bp * Wn + w] = de + sc[1]; nsum += dn; }
  }
  if (lane == 0) nred[wave] = nsum;
  __syncthreads();
  if (threadIdx.x == 0) {
    float s = 0.f;
    for (int i = 0; i < 8; ++i) s += nred[i];
    nc[bp] = s * (1.f / 128.f) + sc[0];
  }
}

// ---------------- K3: lag-correlation edges via WMMA + float atomics -----------
__global__ __launch_bounds__(32) void k_edges(const float* Bp, const float* Cp,
                                              const float* ebuf, float* edges) {
  int b = blockIdx.x, L = blockIdx.y, chunk = blockIdx.z;
  int nvalid = Wn - L;
  int w0 = chunk * 16;
  int w1 = w0 + 16 < nvalid ? w0 + 16 : nvalid;
  int lane = threadIdx.x & 31, nq = lane & 15, hi = lane >> 4;
  float acc[8];
  for (int r = 0; r < 8; ++r) acc[r] = 0.f;
  const float* Ab = Bp + (long)b * Pn * Wn * Dn;
  const float* Cb = Cp + (long)b * Pn * Wn * Dn;
  for (int w = w0; w < w1; ++w) {
    // prefetch next-w rows for this lane's A/B streams (L2 latency hiding)
    if (w + 1 < w1) {
      __builtin_prefetch(Ab + (long)(w + 1) * Dn + (lane & 15) * (long)Wn * Dn, 0, 3);
      __builtin_prefetch(Cb + (long)(w + 1 + L) * Dn + (lane & 15) * (long)Wn * Dn, 0, 3);
    }
    v8f c = vzero();
    c = gemm_nt(Ab + (long)w * Dn, (long)Wn * Dn,
                Cb + (long)(w + L) * Dn, (long)Wn * Dn, 256, c);
    for (int r = 0; r < 8; ++r) {
      int m = r + 8 * hi;
      float ev = ebuf[((long)b * Pn + m) * Wn + w];
      acc[r] += __expf(c[r]) * ev;
    }
  }
  float inv = 1.f / (float)nvalid;
  for (int r = 0; r < 8; ++r) {
    int m = r + 8 * hi;
    atomicAdd(&edges[(((long)b * Pn + m) * Pn + nq) * 4 + L], acc[r] * inv);
  }
}

// ---------------- K4: scores -> conn, lag_off ----------------------------------
__global__ __launch_bounds__(256) void k_scores(const float* edges, const float* nc,
                                                float* conn, int* lag) {
  int idx = blockIdx.x * 256 + threadIdx.x;  // 2048 = B*P*P
  if (idx >= Bn * Pn * Pn) return;
  int b = idx >> 8, rem = idx & 255, p = rem >> 4, q = rem & 15;
  float n = nc[b * Pn + p];
  float best = -1.f; int bl = 0;
  for (int L = 0; L < 4; ++L) {
    float s = 1.f / (1.f + __expf(n + edges[idx * 4 + L]));  // sigmoid(-(nc+edge))
    if (s > best) { best = s; bl = L; }
  }
  bool od = (p != q);
  conn[idx] = (od && best > 0.5f) ? 1.f : 0.f;
  lag[idx] = od ? bl : 0;
}

// ---------------- K5: delay mix + aligned mean-pool ----------------------------
__global__ __launch_bounds__(256) void k_align(const float* fea, const float* Mr,
                                               const float* Md, const float* mix,
                                               const float* conn, const int* lag,
                                               float* alg) {
  int bp = blockIdx.x;
  int b = bp >> 4;
  int t = threadIdx.x;
  __shared__ float wts[128 * 16];
  __shared__ float conn_s[16];
  __shared__ int lag_s[16];
  __shared__ int hc;
  if (t < 16) { conn_s[t] = conn[bp * 16 + t]; lag_s[t] = lag[bp * 16 + t]; }
  if (t == 0) hc = 0;
  __syncthreads();
  if (t < 16 && conn_s[t] > 0.f) atomicOr(&hc, 1);
  if (t < 128) {
    int w = t;
    const float* ml = mix + ((long)bp * Wn + w) * 16;
    float lg[16], mx = -3.0e38f;
    for (int q = 0; q < 16; ++q) {
      lg[q] = (conn_s[q] > 0.f) ? ml[q] : -1.0e30f;
      mx = fmaxf(mx, lg[q]);
    }
    float s = 0.f;
    for (int q = 0; q < 16; ++q) { lg[q] = __expf(lg[q] - mx); s += lg[q]; }
    float inv = 1.f / s;
    for (int q = 0; q < 16; ++q) wts[w * 16 + q] = lg[q] * inv;
  }
  __syncthreads();
  int d = t;
  float acc = 0.f;
  if (hc) {
    if (d < 128) {
      const float* mr = Mr + (long)bp * Wn * 128 + d;
      for (int w = 0; w < 128; ++w) acc += mr[(long)w * 128];
    } else {
      int dd = d - 128;
      for (int w = 0; w < 128; ++w) {
        float s = 0.f;
        for (int q = 0; q < 16; ++q) {
          int wi = w + lag_s[q]; if (wi > 127) wi = 127;
          s += wts[w * 16 + q] * Md[(((long)b * Pn + q) * Wn + wi) * 128 + dd];
        }
        acc += s;
      }
    }
  } else {
    const float* f = fea + (long)bp * Wn * 256 + d;
    for (int w = 0; w < 128; ++w) acc += f[(long)w * 256];
  }
  alg[(long)bp * 256 + d] = acc * (1.f / 128.f);
}

// ---------------- K6: q/k projections via WMMA ---------------------------------
__global__ __launch_bounds__(256) void k_qk(const float* alg,
                                            const float* w_q, const float* b_q,
                                            const float* w_k, const float* b_k,
                                            float* qb, float* kb) {
  __shared__ float As[16 * 256];
  int rt = blockIdx.x;  // 0..7
  async_stage(alg + (long)rt * 16 * 256, As, 1024);
  int wave = threadIdx.x >> 5;
  int job = blockIdx.y * 8 + wave;  // gridDim.y = 4 -> 0..31
  const float* Wm = (job < 16) ? w_q : w_k;
  const float* bias = (job < 16) ? b_q : b_k;
  float* out = (job < 16) ? qb : kb;
  int ncol = (job & 15) * 16;
  v8f c = vzero();
  c = gemm_w(As, 256, Wm + ncol, 256, 256, c);
  int lane = threadIdx.x & 31, n = lane & 15, hi = lane >> 4;
  float bv = bias[ncol + n];
  float* o = out + (long)rt * 16 * 256 + ncol;
  for (int r = 0; r < 8; ++r) o[(long)(r + 8 * hi) * 256 + n] = c[r] + bv;
}

// ---------------- K7: sim (WMMA) + adjacency + tmp = norm_adj @ aligned --------
__global__ __launch_bounds__(32) void k_adj(const float* qb, const float* kb,
                                            const float* pos, const float* conn,
                                            const float* alg, float* tmp) {
  int b = blockIdx.x;
  int lane = threadIdx.x & 31, n = lane & 15, hi = lane >> 4;
  __shared__ float dinv_s[16];
  __shared__ float na_s[256];
  v8f c = vzero();
  c = gemm_nt(qb + (long)b * 16 * 256, 256, kb + (long)b * 16 * 256, 256, 256, c);
  float px = pos[(b * 16 + n) * 2 + 0], py = pos[(b * 16 + n) * 2 + 1];
  float rel[8];
  for (int r = 0; r < 8; ++r) {
    int m = r + 8 * hi;
    float qx = pos[(b * 16 + m) * 2 + 0], qy = pos[(b * 16 + m) * 2 + 1];
    float dx = qx - px, dy = qy - py;
    float dist = sqrtf(dx * dx + dy * dy);
    float sim = c[r] * (1.f / 16.f);
    rel[r] = __expf(-dist) * __expf(sim) * conn[(b * 16 + m) * 16 + n];
  }
  for (int r = 0; r < 8; ++r) {
    // row m lives entirely in one 16-lane group at register r
    float rs = rel[r], mx = rel[r];
    for (int o = 8; o; o >>= 1) {
      rs += __shfl_xor(rs, o, 32);
      mx = fmaxf(mx, __shfl_xor(mx, o, 32));
    }
    float ex = __expf(rel[r] - mx);
    float es = ex;
    for (int o = 8; o; o >>= 1) es += __shfl_xor(es, o, 32);
    float relf = (rs > 0.f) ? (ex / es) : rel[r];
    float dg = relf;
    for (int o = 8; o; o >>= 1) dg += __shfl_xor(dg, o, 32);
    rel[r] = relf;
    if (n == 0) dinv_s[r + 8 * hi] = rsqrtf(dg + 1e-6f);
  }
  __syncthreads();
  for (int r = 0; r < 8; ++r) {
    int m = r + 8 * hi;
    na_s[m * 16 + n] = dinv_s[m] * rel[r] * dinv_s[n];
  }
  __syncthreads();
  for (int nt = 0; nt < 16; ++nt) {
    v8f t = vzero();
    t = gemm_w(na_s, 16, alg + (long)b * 16 * 256 + nt * 16, 256, 16, t);
    float* o = tmp + (long)b * 16 * 256 + nt * 16;
    for (int r = 0; r < 8; ++r) o[(long)(r + 8 * hi) * 256 + n] = t[r];
  }
}

// ---------------- K8: out = relu(tmp @ w_g) ------------------------------------
__global__ __launch_bounds__(256) void k_out(const float* tmp, const float* w_g,
                                             float* out) {
  __shared__ float As[16 * 256];
  int rt = blockIdx.x;  // 0..7
  async_stage(tmp + (long)rt * 16 * 256, As, 1024);
  int wave = threadIdx.x >> 5;
  int nt = blockIdx.y * 8 + wave;  // gridDim.y = 2 -> 0..15
  v8f c = vzero();
  c = gemm_w(As, 256, w_g + nt * 16, 256, 256, c);
  int lane = threadIdx.x & 31, n = lane & 15, hi = lane >> 4;
  float* o = out + (long)rt * 16 * 256 + nt * 16;
  for (int r = 0; r < 8; ++r)
    o[(long)(r + 8 * hi) * 256 + n] = fmaxf(c[r], 0.f);
}

// ---------------- workspace layout (floats) ------------------------------------
constexpr long O_BP = 0;
constexpr long O_CP = O_BP + (long)Bn * Pn * Wn * Dn;           // 4,194,304
constexpr long O_MR = O_CP + (long)Bn * Pn * Wn * Dn;
constexpr long O_MD = O_MR + (long)Bn * Pn * Wn * 128;
constexpr long O_SM = O_MD + (long)Bn * Pn * Wn * 128;
constexpr long O_WN = O_SM;
constexpr long O_VAE = O_WN + 256;
constexpr long O_SC = O_VAE + 256;
constexpr long O_NC = O_SC + 16;
constexpr long O_EDG = O_NC + 128;
constexpr long O_CONN = O_EDG + (long)Bn * Pn * Pn * 4;
constexpr long O_LAG = O_CONN + (long)Bn * Pn * Pn;
constexpr long O_EB = O_LAG + (long)Bn * Pn * Pn;
constexpr long O_ALG = O_EB + (long)Bn * Pn * Wn;
constexpr long O_QB = O_ALG + (long)Bn * Pn * Dn;
constexpr long O_KB = O_QB + (long)Bn * Pn * Dn;
constexpr long O_TMP = O_KB + (long)Bn * Pn * Dn;

extern "C" void kernel_launch(void* const* d_in, const int* in_sizes, int n_in,
                              void* d_out, int out_size, void* d_ws, size_t ws_size,
                              hipStream_t stream) {
  (void)in_sizes; (void)n_in; (void)out_size; (void)ws_size;
  const float* fea    = (const float*)d_in[0];
  const float* pos    = (const float*)d_in[1];
  const float* mix    = (const float*)d_in[2];
  const float* w_node = (const float*)d_in[3];
  const float* b_node = (const float*)d_in[4];
  const float* w_a    = (const float*)d_in[5];
  const float* b_a    = (const float*)d_in[6];
  const float* w_b    = (const float*)d_in[7];
  const float* b_b    = (const float*)d_in[8];
  const float* w_c    = (const float*)d_in[9];
  const float* b_c    = (const float*)d_in[10];
  const float* w_edge = (const float*)d_in[11];
  const float* b_edge = (const float*)d_in[12];
  const float* w_real = (const float*)d_in[13];
  const float* b_real = (const float*)d_in[14];
  const float* w_del  = (const float*)d_in[15];
  const float* b_del  = (const float*)d_in[16];
  const float* w_q    = (const float*)d_in[17];
  const float* b_q    = (const float*)d_in[18];
  const float* w_k    = (const float*)d_in[19];
  const float* b_k    = (const float*)d_in[20];
  const float* w_g    = (const float*)d_in[21];
  float* ws = (float*)d_ws;
  float* out = (float*)d_out;

  k_prep<<<1, 256, 0, stream>>>(w_node, b_node, w_a, b_a, w_edge, b_edge,
                                ws + O_WN, ws + O_VAE, ws + O_SC, ws + O_EDG);
  k_proj<<<dim3(1024, 6), 256, 0, stream>>>(fea, w_b, b_b, w_c, b_c, w_real,
                                            b_real, w_del, b_del, ws + O_BP,
                                            ws + O_CP, ws + O_MR, ws + O_MD);
  k_e_node<<<128, 256, 0, stream>>>(fea, ws + O_WN, ws + O_VAE, ws + O_SC,
                                    ws + O_EB, ws + O_NC);
  k_edges<<<dim3(8, 4, 8), 32, 0, stream>>>(ws + O_BP, ws + O_CP, ws + O_EB,
                                            ws + O_EDG);
  k_scores<<<8, 256, 0, stream>>>(ws + O_EDG, ws + O_NC, ws + O_CONN,
                                  (int*)(ws + O_LAG));
  k_align<<<128, 256, 0, stream>>>(fea, ws + O_MR, ws + O_MD, mix, ws + O_CONN,
                                   (const int*)(ws + O_LAG), ws + O_ALG);
  k_qk<<<dim3(8, 4), 256, 0, stream>>>(ws + O_ALG, w_q, b_q, w_k, b_k,
                                       ws + O_QB, ws + O_KB);
  k_adj<<<8, 32, 0, stream>>>(ws + O_QB, ws + O_KB, pos, ws + O_CONN,
                              ws + O_ALG, ws + O_TMP);
  k_out<<<dim3(8, 2), 256, 0, stream>>>(ws + O_TMP, w_g, out);
}